// NeuralNetwork_63556926046364
// MI455X (gfx1250) — compile-verified
//
#include <hip/hip_runtime.h>
#include <math.h>

// ---------------------------------------------------------------------------
// NeuralNetwork chain: 24 layers of out = silu((W .* M) @ v + b), last layer without silu, fp32.
// HBM-bound: 805 MB of W+M streamed once -> ~35us floor at 23.3 TB/s.
// One row per wave32; coalesced b128 NT loads; WMMA f32 16x16x4 used for the
// exact fp32 cross-lane reduction of the dot-product partials.
// ---------------------------------------------------------------------------

#define WIDTH        2048
#define N_LAYERS     24
#define ROWS_PER_BLK 8
#define THREADS      (ROWS_PER_BLK * 32)   // 8 waves of 32

typedef __attribute__((ext_vector_type(2))) float v2f;
typedef __attribute__((ext_vector_type(4))) float vf4;
typedef __attribute__((ext_vector_type(8))) float v8f;

__global__ __launch_bounds__(256) void init_values_kernel(
    const float* __restrict__ x, float* __restrict__ values)
{
    int i = blockIdx.x * blockDim.x + threadIdx.x;
    if (i < WIDTH) values[i] = x[i];
}

__global__ __launch_bounds__(THREADS) void layer_kernel(
    const float* __restrict__ W,      // [WIDTH, WIDTH] this layer's weights
    const float* __restrict__ M,      // [WIDTH, WIDTH] this layer's mask
    const float* __restrict__ bias,   // [WIDTH]
    const int*   __restrict__ idx,    // indices[l], WIDTH ints (gather ids)
    const float* __restrict__ values, // full flat neuron-value buffer (read)
    float*       __restrict__ dst,    // values buffer, or d_out for last layer
    int dst_sub,                      // 0, or N_LAYERS*WIDTH for last layer
    int apply_silu)
{
    // Stage the gathered activation vector in LDS once per block (8 KB).
    __shared__ vf4 vals4[WIDTH / 4];
    {
        float* vals = (float*)vals4;
        for (int j = threadIdx.x; j < WIDTH; j += THREADS)
            vals[j] = values[idx[j]];
    }
    __syncthreads();

    const int lane = threadIdx.x & 31;
    const int wave = threadIdx.x >> 5;
    const int row  = blockIdx.x * ROWS_PER_BLK + wave;

    const vf4* __restrict__ Wrow = (const vf4*)(W + (size_t)row * WIDTH);
    const vf4* __restrict__ Mrow = (const vf4*)(M + (size_t)row * WIDTH);

    // Each lane streams 16 float4s of this row: fully coalesced 512B/instr,
    // non-temporal (dead after use -> don't pollute L2). Unroll x8 keeps
    // ~16 b128 loads in flight per wave to cover HBM latency.
    vf4 acc = {0.0f, 0.0f, 0.0f, 0.0f};
#pragma unroll 8
    for (int t = 0; t < WIDTH / 128; ++t) {
        int c = t * 32 + lane;                         // float4 column index
        vf4 w = __builtin_nontemporal_load(&Wrow[c]);
        vf4 m = __builtin_nontemporal_load(&Mrow[c]);
        vf4 v = vals4[c];
        acc.x = fmaf(w.x * m.x, v.x, acc.x);
        acc.y = fmaf(w.y * m.y, v.y, acc.y);
        acc.z = fmaf(w.z * m.z, v.z, acc.z);
        acc.w = fmaf(w.w * m.w, v.w, acc.w);
    }
    float p = (acc.x + acc.y) + (acc.z + acc.w);       // per-lane partial

    // Exact fp32 32-lane reduction via V_WMMA_F32_16X16X4_F32.
    // A(16x4): lane L<16 -> A[L][0]=p, A[L][1]=0 ; lane L>=16 -> A[L-16][2]=p,
    // A[L-16][3]=0. B = all-ones 4x16. => D[i][j] = p_i + p_{i+16} for all j.
    // Uniform control flow + full blocks => EXEC is all ones here (required).
    v2f a;  a[0] = p;    a[1] = 0.0f;
    v2f bm; bm[0] = 1.0f; bm[1] = 1.0f;
    v8f cz = {};
    v8f d = __builtin_amdgcn_wmma_f32_16x16x4_f32(
        /*neg_a=*/false, a, /*neg_b=*/false, bm,
        /*c_mod=*/(short)0, cz, /*reuse_a=*/false, /*reuse_b=*/false);

    // C/D layout: VGPR r holds row r (lanes 0-15) and row r+8 (lanes 16-31).
    // Lanes <16 sum rows 0..7, lanes >=16 sum rows 8..15; one shfl finishes.
    float t8 = ((d[0] + d[1]) + (d[2] + d[3])) + ((d[4] + d[5]) + (d[6] + d[7]));
    float total = t8 + __shfl_xor(t8, 16, 32);         // full 32-lane sum

    if (lane == 0) {
        float y = total + bias[row];
        if (apply_silu) y = y / (1.0f + expf(-y));     // silu(y) = y*sigmoid(y)
        int out_id = idx[row] + WIDTH - dst_sub;       // scatter target
        dst[out_id] = y;
    }
}

extern "C" void kernel_launch(void* const* d_in, const int* in_sizes, int n_in,
                              void* d_out, int out_size, void* d_ws, size_t ws_size,
                              hipStream_t stream)
{
    const float* x       = (const float*)d_in[0];
    const float* weights = (const float*)d_in[1];
    const float* masks   = (const float*)d_in[2];
    const float* biases  = (const float*)d_in[3];
    const int*   indices = (const int*)d_in[4];
    float*       out     = (float*)d_out;

    // Flat neuron-value buffer, (N_LAYERS+1)*WIDTH floats = 200 KB of d_ws.
    float* values = (float*)d_ws;

    init_values_kernel<<<(WIDTH + 255) / 256, 256, 0, stream>>>(x, values);

    for (int l = 0; l < N_LAYERS; ++l) {
        const float* Wl = weights + (size_t)l * WIDTH * WIDTH;
        const float* Ml = masks   + (size_t)l * WIDTH * WIDTH;
        const float* bl = biases  + (size_t)l * WIDTH;
        const int*   il = indices + (size_t)l * WIDTH;
        const bool last = (l == N_LAYERS - 1);

        layer_kernel<<<WIDTH / ROWS_PER_BLK, THREADS, 0, stream>>>(
            Wl, Ml, bl, il, values,
            last ? out : values,
            last ? N_LAYERS * WIDTH : 0,
            last ? 0 : 1);
    }
}